// PolarPredictor_55035710931322
// MI455X (gfx1250) — compile-verified
//
#include <hip/hip_runtime.h>
#include <hip/hip_bf16.h>

typedef __attribute__((ext_vector_type(2))) float v2f;
typedef __attribute__((ext_vector_type(4))) float v4f;
typedef __attribute__((ext_vector_type(8))) float v8f;

#define HH 96
#define WW 96
#define HW (HH * WW)          // 9216
#define CIN 256
#define OCH 96                // 3*NUM_CHARS
#define NCH 32                // NUM_CHARS
#define VOC 38
#define VPAD 48
#define PTILES (HW / 16)      // 576

#define RECS_SZ ((size_t)VOC * HW * NCH)   // 11,206,656
#define MASKS_SZ ((size_t)NCH * HW)        // 294,912

// ws layout (floats)
#define FEATS_OFF 0
#define Z_OFF     ((size_t)OCH * HW)                 // 884,736
#define WT_OFF    (Z_OFF + (size_t)HW * VPAD)        // +442,368
#define WCT_OFF   (WT_OFF + (size_t)9 * CIN * OCH)   // +221,184
// total: 1,560,576 floats ≈ 6.2 MB

// ---------------------------------------------------------------------------
// Prep 1: wT2[tap][ic/2][oc][2] = w_loc[oc][ic][tap]
// Pair-interleaved K so an A fragment is ONE b64 load per lane (coalesced).
// ---------------------------------------------------------------------------
__global__ __launch_bounds__(256) void k_wt(const float* __restrict__ w,
                                            float* __restrict__ wT2) {
  const int idx = blockIdx.x * 256 + threadIdx.x;   // 9*256*96 = 221184
  if (idx >= 9 * CIN * OCH) return;
  const int tap = idx / (CIN * OCH);
  const int ic  = (idx / OCH) % CIN;
  const int oc  = idx % OCH;
  wT2[(((size_t)tap * (CIN / 2) + (ic >> 1)) * OCH + oc) * 2 + (ic & 1)] =
      w[((size_t)oc * CIN + ic) * 9 + tap];
}

// ---------------------------------------------------------------------------
// Prep 2: wclfT2[ic/2][v][2] = w_clf[v][ic], v zero-padded to 48
// ---------------------------------------------------------------------------
__global__ __launch_bounds__(256) void k_wct(const float* __restrict__ wclf,
                                             float* __restrict__ wclfT2) {
  const int idx = blockIdx.x * 256 + threadIdx.x;   // 256*48 = 12288
  if (idx >= CIN * VPAD) return;
  const int ic = idx / VPAD;
  const int v  = idx % VPAD;
  wclfT2[(((size_t)(ic >> 1)) * VPAD + v) * 2 + (ic & 1)] =
      (v < VOC) ? wclf[(size_t)v * CIN + ic] : 0.0f;
}

// ---------------------------------------------------------------------------
// Kernel 1: 3x3 conv (pad=1), implicit GEMM via V_WMMA_F32_16X16X4_F32.
// Block = 9 waves, one tap per wave (K-split for occupancy: 5184 waves);
// each wave computes 16 px x 96 oc (6 acc tiles, B reused 6x, A = b64);
// tap partials reduced through LDS (55 KB/block, WGP has 320 KB).
// ---------------------------------------------------------------------------
__global__ __launch_bounds__(288) void k_conv(const float* __restrict__ x,
                                              const float* __restrict__ wT2,
                                              const float* __restrict__ bias,
                                              float* __restrict__ feats) {
  __shared__ float lds[9][6][8][32];
  const int lane = threadIdx.x & 31;
  const int tap  = threadIdx.x >> 5;           // 0..8
  const int n  = lane & 15;
  const int hi = lane >> 4;
  const int h  = blockIdx.x / (WW / 16);
  const int w0 = (blockIdx.x % (WW / 16)) * 16;

  const int ky = tap / 3 - 1;
  const int kx = tap % 3 - 1;
  const int py = h + ky;
  const int px = w0 + n + kx;
  const bool valid = (py >= 0) && (py < HH) && (px >= 0) && (px < WW);
  const int xoff = valid ? py * WW + px : 0;
  const float vm = valid ? 1.0f : 0.0f;
  const float* wtap = wT2 + (size_t)tap * (CIN / 2) * OCH * 2;

  v8f acc[6] = {};
  for (int ic0 = 0; ic0 < CIN; ic0 += 4) {
    const int kk = ic0 + 2 * hi;               // even
    v2f b;
    b.x = vm * x[(size_t)(kk + 0) * HW + xoff];
    b.y = vm * x[(size_t)(kk + 1) * HW + xoff];
    const float* abase = wtap + ((size_t)(kk >> 1) * OCH + n) * 2;
#pragma unroll
    for (int t = 0; t < 6; ++t) {
      const v2f a = *(const v2f*)(abase + t * 32);   // b64, lane-contiguous
      acc[t] = __builtin_amdgcn_wmma_f32_16x16x4_f32(
          false, a, false, b, (short)0, acc[t], false, false);
    }
  }

#pragma unroll
  for (int t = 0; t < 6; ++t)
#pragma unroll
    for (int r = 0; r < 8; ++r)
      lds[tap][t][r][lane] = acc[t][r];
  __syncthreads();

  // reduce 9 tap-partials: 6*8*32 = 1536 values over 288 threads
  for (int j = threadIdx.x; j < 6 * 8 * 32; j += 288) {
    const int l = j & 31;
    const int r = (j >> 5) & 7;
    const int t = j >> 8;
    float s = 0.0f;
#pragma unroll
    for (int wv = 0; wv < 9; ++wv) s += lds[wv][t][r][l];
    const int oc = t * 16 + r + 8 * (l >> 4);  // C/D layout: vgpr r -> r+8*hi
    const int p  = h * WW + w0 + (l & 15);
    feats[(size_t)oc * HW + p] = s + bias[oc];
  }
}

// ---------------------------------------------------------------------------
// Kernel 2: z[p][v] = sum_c w_clf[v][c] * y[c][p].  Block = 4 waves, each
// covering 64 channels (K-split: 2304 waves), 3 acc tiles/wave, LDS reduce.
// ---------------------------------------------------------------------------
__global__ __launch_bounds__(128) void k_proj(const float* __restrict__ y,
                                              const float* __restrict__ wclfT2,
                                              float* __restrict__ z) {
  __shared__ float lds[4][3][8][32];
  const int lane = threadIdx.x & 31;
  const int wv   = threadIdx.x >> 5;           // 0..3
  const int n  = lane & 15;
  const int hi = lane >> 4;
  const int p0 = blockIdx.x * 16;

  v8f acc[3] = {};
  for (int ic0 = wv * 64; ic0 < wv * 64 + 64; ic0 += 4) {
    const int kk = ic0 + 2 * hi;
    v2f b;
    b.x = y[(size_t)(kk + 0) * HW + p0 + n];
    b.y = y[(size_t)(kk + 1) * HW + p0 + n];
    const float* abase = wclfT2 + ((size_t)(kk >> 1) * VPAD + n) * 2;
#pragma unroll
    for (int t = 0; t < 3; ++t) {
      const v2f a = *(const v2f*)(abase + t * 32);
      acc[t] = __builtin_amdgcn_wmma_f32_16x16x4_f32(
          false, a, false, b, (short)0, acc[t], false, false);
    }
  }

#pragma unroll
  for (int t = 0; t < 3; ++t)
#pragma unroll
    for (int r = 0; r < 8; ++r)
      lds[wv][t][r][lane] = acc[t][r];
  __syncthreads();

  // reduce 4 partials: 3*8*32 = 768 values over 128 threads
  for (int j = threadIdx.x; j < 3 * 8 * 32; j += 128) {
    const int l = j & 31;
    const int r = (j >> 5) & 7;
    const int t = j >> 8;
    float s = lds[0][t][r][l] + lds[1][t][r][l] +
              lds[2][t][r][l] + lds[3][t][r][l];
    const int v = t * 16 + r + 8 * (l >> 4);
    const int p = p0 + (l & 15);
    z[(size_t)p * VPAD + v] = s;               // pixel-major, padded to 48
  }
}

// ---------------------------------------------------------------------------
// Kernel 3a: cumsum over chars -> locs_cat; copy masks. One thread per pixel.
// ---------------------------------------------------------------------------
__global__ __launch_bounds__(256) void k_locs(const float* __restrict__ feats,
                                              float* __restrict__ out) {
  const int p = blockIdx.x * blockDim.x + threadIdx.x;
  if (p >= HW) return;
  float* masks_out = out + RECS_SZ;
  float* locs_out  = out + RECS_SZ + MASKS_SZ;
  float sy = 0.0f, sx = 0.0f;
  for (int k = 0; k < NCH; ++k) {
    sy += feats[(size_t)(2 * k + 0) * HW + p];
    sx += feats[(size_t)(2 * k + 1) * HW + p];
    locs_out[(size_t)(2 * k + 0) * HW + p] = sy;
    locs_out[(size_t)(2 * k + 1) * HW + p] = sx;
    masks_out[(size_t)k * HW + p] = feats[(size_t)(2 * NCH + k) * HW + p];
  }
}

// ---------------------------------------------------------------------------
// Kernel 3b: bilinear sample of projected z + bias -> recs [v][h][w][k].
// One wave per pixel, lane = char k. Taps read as float4 (global_load_b128);
// per-v stores are 128B coalesced lines.
// ---------------------------------------------------------------------------
__global__ __launch_bounds__(256) void k_sample(const float* __restrict__ z,
                                                const float* __restrict__ bclf,
                                                const float* __restrict__ locs,
                                                float* __restrict__ out) {
  const int k    = threadIdx.x & 31;
  const int wave = threadIdx.x >> 5;
  const int p = blockIdx.x * 8 + wave;
  const int hh = p / WW;
  const int wc = p % WW;

  const float py = locs[(size_t)(2 * k + 0) * HW + p] + (float)hh;
  const float px = locs[(size_t)(2 * k + 1) * HW + p] + (float)wc;
  const float y0f = floorf(py);
  const float x0f = floorf(px);
  const int y0 = (int)y0f, x0 = (int)x0f;
  const int y1 = y0 + 1,   x1 = x0 + 1;
  const float wy = py - y0f, wx = px - x0f;

  const float vy0 = (y0 >= 0 && y0 < HH) ? 1.0f : 0.0f;
  const float vy1 = (y1 >= 0 && y1 < HH) ? 1.0f : 0.0f;
  const float vx0 = (x0 >= 0 && x0 < WW) ? 1.0f : 0.0f;
  const float vx1 = (x1 >= 0 && x1 < WW) ? 1.0f : 0.0f;
  const int y0c = min(max(y0, 0), HH - 1);
  const int y1c = min(max(y1, 0), HH - 1);
  const int x0c = min(max(x0, 0), WW - 1);
  const int x1c = min(max(x1, 0), WW - 1);

  const float w00 = (1.0f - wy) * (1.0f - wx) * vy0 * vx0;
  const float w01 = (1.0f - wy) * wx          * vy0 * vx1;
  const float w10 = wy          * (1.0f - wx) * vy1 * vx0;
  const float w11 = wy          * wx          * vy1 * vx1;

  const v4f* z00 = (const v4f*)(z + (size_t)(y0c * WW + x0c) * VPAD);
  const v4f* z01 = (const v4f*)(z + (size_t)(y0c * WW + x1c) * VPAD);
  const v4f* z10 = (const v4f*)(z + (size_t)(y1c * WW + x0c) * VPAD);
  const v4f* z11 = (const v4f*)(z + (size_t)(y1c * WW + x1c) * VPAD);

  float* orow = out + (size_t)hh * WW * NCH + (size_t)wc * NCH + k;
#pragma unroll
  for (int c = 0; c < 10; ++c) {        // 10*4 = 40 >= 38 values
    const v4f val = w00 * z00[c] + w01 * z01[c] +
                    w10 * z10[c] + w11 * z11[c];
#pragma unroll
    for (int e = 0; e < 4; ++e) {
      const int v = c * 4 + e;
      if (v < VOC) {
        orow[(size_t)v * HW * NCH] = val[e] + bclf[v];
      }
    }
  }
}

extern "C" void kernel_launch(void* const* d_in, const int* in_sizes, int n_in,
                              void* d_out, int out_size, void* d_ws, size_t ws_size,
                              hipStream_t stream) {
  const float* x     = (const float*)d_in[0];
  const float* y     = (const float*)d_in[1];
  const float* w_loc = (const float*)d_in[2];
  const float* b_loc = (const float*)d_in[3];
  const float* w_clf = (const float*)d_in[4];
  const float* b_clf = (const float*)d_in[5];
  float* out = (float*)d_out;

  float* ws     = (float*)d_ws;
  float* feats  = ws + FEATS_OFF;    // [96][9216]
  float* z      = ws + Z_OFF;        // [9216][48]
  float* wT2    = ws + WT_OFF;       // [9][128][96][2]
  float* wclfT2 = ws + WCT_OFF;      // [128][48][2]

  k_wt <<<(9 * CIN * OCH + 255) / 256, 256, 0, stream>>>(w_loc, wT2);
  k_wct<<<(CIN * VPAD + 255) / 256, 256, 0, stream>>>(w_clf, wclfT2);
  k_conv<<<PTILES, 288, 0, stream>>>(x, wT2, b_loc, feats);
  k_proj<<<PTILES, 128, 0, stream>>>(y, wclfT2, z);
  k_locs<<<HW / 256, 256, 0, stream>>>(feats, out);
  const float* locs = out + RECS_SZ + MASKS_SZ;
  k_sample<<<HW / 8, 256, 0, stream>>>(z, b_clf, locs, out);
}